// Generator3DLUT_identity_20830591385730
// MI455X (gfx1250) — compile-verified
//
#include <hip/hip_runtime.h>

// 3D LUT trilinear interpolation, MI455X (gfx1250, wave32).
// Roofline: 192 MB HBM stream @ 23.3 TB/s ~ 8.3 us floor; ~2.5 flop/byte ->
// far below machine balance, so no WMMA (intentional; no matmul structure).
// gfx1250-specific choices:
//   * float4-packed LUT -> each corner gather is ONE global_load_b128
//   * x loads / out stores use TH=NT (th:TH_NT) so the 2x96 MB stream does
//     not evict the 575 KB LUT from WGP$/L2 (CDNA5 ISA sec 7.3)
//   * global_prefetch_b8 on the input stream
//   * wave32 blocks of 256 (8 waves), grid exactly covers the domain

#define DIM   33
#define NLUT  (DIM * DIM * DIM)          // 35937 grid points
#define BATCH 8
#define PIX   (1024 * 1024)              // pixels per (batch, channel) plane
#define GROUPS_PER_BATCH (PIX / 4)       // 262144 (2^18)
#define TOTAL_GROUPS (BATCH * GROUPS_PER_BATCH)

typedef __attribute__((ext_vector_type(4))) float v4f;

__device__ __forceinline__ float lerpf(float a, float b, float t) {
    return fmaf(t, b - a, a);            // (1-t)*a + t*b
}

// ---- Pass 1: repack planar LUT (3,D,D,D) -> interleaved v4f[NLUT] ----------
__global__ __launch_bounds__(256) void pack_lut_kernel(
    const float* __restrict__ lut, v4f* __restrict__ packed)
{
    int i = blockIdx.x * 256 + threadIdx.x;
    if (i < NLUT) {
        v4f v;
        v.x = lut[i];                    // channel 0 plane
        v.y = lut[i + NLUT];             // channel 1 plane
        v.z = lut[i + 2 * NLUT];         // channel 2 plane
        v.w = 0.0f;
        packed[i] = v;                   // regular (temporal) store: reused by pass 2
    }
}

// ---- Pass 2: apply LUT, packed path (8 x global_load_b128 gathers/pixel) ---
__global__ __launch_bounds__(256) void apply_lut_packed_kernel(
    const float* __restrict__ x, const v4f* __restrict__ lut,
    float* __restrict__ out)
{
    const int tid   = blockIdx.x * 256 + threadIdx.x;      // grid exactly covers TOTAL_GROUPS
    const int batch = tid >> 18;                           // / GROUPS_PER_BATCH
    const int p     = (tid & (GROUPS_PER_BATCH - 1)) << 2; // 4 pixels per thread

    const float* xb = x   + (size_t)batch * 3 * PIX;
    float*       ob = out + (size_t)batch * 3 * PIX;

    // Speculative stream prefetch (-> global_prefetch_b8; safe if OOB).
    __builtin_prefetch(xb + p + 2048, 0, 0);

    // Streaming inputs: non-temporal (th:TH_NT) -- touched once, keep LUT hot.
    const v4f xr4 = __builtin_nontemporal_load((const v4f*)(xb + p));           // ch 0 (r)
    const v4f xg4 = __builtin_nontemporal_load((const v4f*)(xb + PIX + p));     // ch 1 (g)
    const v4f xb4 = __builtin_nontemporal_load((const v4f*)(xb + 2 * PIX + p)); // ch 2 (b)

    const float rr[4] = {xr4.x, xr4.y, xr4.z, xr4.w};
    const float gg[4] = {xg4.x, xg4.y, xg4.z, xg4.w};
    const float bb[4] = {xb4.x, xb4.y, xb4.z, xb4.w};
    float orr[4], ogg[4], obb[4];

    const float scale = (float)(DIM - 1) / 1.0001f;        // 1/binsize

#pragma unroll
    for (int i = 0; i < 4; ++i) {
        const float sr = rr[i] * scale;
        const float sg = gg[i] * scale;
        const float sb = bb[i] * scale;
        const float fr = floorf(sr), fg = floorf(sg), fb = floorf(sb);
        const float dr = sr - fr, dg = sg - fg, db = sb - fb;
        const int r0 = (int)fr, g0 = (int)fg, b0 = (int)fb;

        const int base = (b0 * DIM + g0) * DIM + r0;       // in-bounds: idx<=31, +1<=32
        // LUT gathers keep default temporal policy (we WANT these cached).
        const v4f c000 = lut[base];
        const v4f c100 = lut[base + 1];
        const v4f c010 = lut[base + DIM];
        const v4f c110 = lut[base + DIM + 1];
        const v4f c001 = lut[base + DIM * DIM];
        const v4f c101 = lut[base + DIM * DIM + 1];
        const v4f c011 = lut[base + DIM * DIM + DIM];
        const v4f c111 = lut[base + DIM * DIM + DIM + 1];

        // lerp along r
        const float x00 = lerpf(c000.x, c100.x, dr), y00 = lerpf(c000.y, c100.y, dr), z00 = lerpf(c000.z, c100.z, dr);
        const float x10 = lerpf(c010.x, c110.x, dr), y10 = lerpf(c010.y, c110.y, dr), z10 = lerpf(c010.z, c110.z, dr);
        const float x01 = lerpf(c001.x, c101.x, dr), y01 = lerpf(c001.y, c101.y, dr), z01 = lerpf(c001.z, c101.z, dr);
        const float x11 = lerpf(c011.x, c111.x, dr), y11 = lerpf(c011.y, c111.y, dr), z11 = lerpf(c011.z, c111.z, dr);
        // lerp along g
        const float x0 = lerpf(x00, x10, dg), y0 = lerpf(y00, y10, dg), z0 = lerpf(z00, z10, dg);
        const float x1 = lerpf(x01, x11, dg), y1 = lerpf(y01, y11, dg), z1 = lerpf(z01, z11, dg);
        // lerp along b
        orr[i] = lerpf(x0, x1, db);
        ogg[i] = lerpf(y0, y1, db);
        obb[i] = lerpf(z0, z1, db);
    }

    // Streaming outputs: non-temporal stores (write-once).
    const v4f vr = {orr[0], orr[1], orr[2], orr[3]};
    const v4f vg = {ogg[0], ogg[1], ogg[2], ogg[3]};
    const v4f vb = {obb[0], obb[1], obb[2], obb[3]};
    __builtin_nontemporal_store(vr, (v4f*)(ob + p));
    __builtin_nontemporal_store(vg, (v4f*)(ob + PIX + p));
    __builtin_nontemporal_store(vb, (v4f*)(ob + 2 * PIX + p));
}

// ---- Fallback: gather directly from planar LUT (3 scalar loads per corner) -
__global__ __launch_bounds__(256) void apply_lut_direct_kernel(
    const float* __restrict__ x, const float* __restrict__ lut,
    float* __restrict__ out)
{
    const int tid   = blockIdx.x * 256 + threadIdx.x;
    const int batch = tid >> 18;
    const int p     = (tid & (GROUPS_PER_BATCH - 1)) << 2;

    const float* xb = x   + (size_t)batch * 3 * PIX;
    float*       ob = out + (size_t)batch * 3 * PIX;

    const v4f xr4 = __builtin_nontemporal_load((const v4f*)(xb + p));
    const v4f xg4 = __builtin_nontemporal_load((const v4f*)(xb + PIX + p));
    const v4f xb4 = __builtin_nontemporal_load((const v4f*)(xb + 2 * PIX + p));

    const float rr[4] = {xr4.x, xr4.y, xr4.z, xr4.w};
    const float gg[4] = {xg4.x, xg4.y, xg4.z, xg4.w};
    const float bb[4] = {xb4.x, xb4.y, xb4.z, xb4.w};
    float orr[4], ogg[4], obb[4];

    const float scale = (float)(DIM - 1) / 1.0001f;

#pragma unroll
    for (int i = 0; i < 4; ++i) {
        const float sr = rr[i] * scale;
        const float sg = gg[i] * scale;
        const float sb = bb[i] * scale;
        const float fr = floorf(sr), fg = floorf(sg), fb = floorf(sb);
        const float dr = sr - fr, dg = sg - fg, db = sb - fb;
        const int r0 = (int)fr, g0 = (int)fg, b0 = (int)fb;
        const int base = (b0 * DIM + g0) * DIM + r0;

        float res[3];
#pragma unroll
        for (int c = 0; c < 3; ++c) {
            const float* L = lut + c * NLUT + base;
            const float x00 = lerpf(L[0],               L[1],                   dr);
            const float x10 = lerpf(L[DIM],             L[DIM + 1],             dr);
            const float x01 = lerpf(L[DIM * DIM],       L[DIM * DIM + 1],       dr);
            const float x11 = lerpf(L[DIM * DIM + DIM], L[DIM * DIM + DIM + 1], dr);
            const float x0  = lerpf(x00, x10, dg);
            const float x1  = lerpf(x01, x11, dg);
            res[c] = lerpf(x0, x1, db);
        }
        orr[i] = res[0]; ogg[i] = res[1]; obb[i] = res[2];
    }

    const v4f vr = {orr[0], orr[1], orr[2], orr[3]};
    const v4f vg = {ogg[0], ogg[1], ogg[2], ogg[3]};
    const v4f vb = {obb[0], obb[1], obb[2], obb[3]};
    __builtin_nontemporal_store(vr, (v4f*)(ob + p));
    __builtin_nontemporal_store(vg, (v4f*)(ob + PIX + p));
    __builtin_nontemporal_store(vb, (v4f*)(ob + 2 * PIX + p));
}

extern "C" void kernel_launch(void* const* d_in, const int* in_sizes, int n_in,
                              void* d_out, int out_size, void* d_ws, size_t ws_size,
                              hipStream_t stream) {
    const float* lut = (const float*)d_in[0];   // (3, 33, 33, 33) fp32
    const float* x   = (const float*)d_in[1];   // (8, 3, 1024, 1024) fp32
    float*       out = (float*)d_out;           // (8, 3, 1024, 1024) fp32
    (void)in_sizes; (void)n_in; (void)out_size;

    const size_t packed_bytes = (size_t)NLUT * sizeof(v4f);     // ~575 KB
    const int grid = TOTAL_GROUPS / 256;                        // 8192, exact cover

    if (ws_size >= packed_bytes) {
        pack_lut_kernel<<<(NLUT + 255) / 256, 256, 0, stream>>>(lut, (v4f*)d_ws);
        apply_lut_packed_kernel<<<grid, 256, 0, stream>>>(x, (const v4f*)d_ws, out);
    } else {
        apply_lut_direct_kernel<<<grid, 256, 0, stream>>>(x, lut, out);
    }
}